// DynamicAttention_7000796693045
// MI455X (gfx1250) — compile-verified
//
#include <hip/hip_runtime.h>
#include <cmath>

typedef __attribute__((ext_vector_type(2))) float v2f;
typedef __attribute__((ext_vector_type(8))) float v8f;

#define B_ 4
#define L_ 1024
#define C_ 128
#define D_ 64
#define NEG_SLOPE 0.01f
// leaky(t) = P*t + Q*|t|
#define P_COEF (0.5f * (1.0f + NEG_SLOPE))
#define Q_COEF (0.5f * (1.0f - NEG_SLOPE))

// ---------------------------------------------------------------------------
// Stage 1: g_l = Q @ Wl, g_r = Q @ Wr   (fp32 WMMA 16x16x4)
// One wave per 16(row) x 16(d) tile; both outputs share the A operand.
// A layout (16x4 f32): lane l, vgpr v  ->  A[m = l&15][k = (l>>4)*2 + v]
// B layout (4x16 f32): lane l, vgpr v  ->  B[k = (l>>4)*2 + v][n = l&15]
// D layout (16x16 f32): vgpr v, lane l ->  D[m = v + (l>>4)*8][n = l&15]
// ---------------------------------------------------------------------------
__global__ void proj_kernel(const float* __restrict__ q,   // (B*L, C)
                            const float* __restrict__ wl,  // (C, D)
                            const float* __restrict__ wr,  // (C, D)
                            float* __restrict__ gl,        // (B*L, D)
                            float* __restrict__ gr)        // (B*L, D)
{
    const int tile = blockIdx.x;
    const int dt   = tile % (D_ / 16);
    const int rt   = tile / (D_ / 16);
    const int row0 = rt * 16;
    const int col0 = dt * 16;
    const int lane = threadIdx.x;
    const int m    = lane & 15;
    const int half = lane >> 4;

    v8f accL = {};
    v8f accR = {};

#pragma unroll 4
    for (int k0 = 0; k0 < C_; k0 += 4) {
        const int k = k0 + half * 2;
        const float* ap = q + (size_t)(row0 + m) * C_ + k;
        v2f a; a.x = ap[0]; a.y = ap[1];
        const float* blp = wl + (size_t)k * D_ + col0 + m;
        const float* brp = wr + (size_t)k * D_ + col0 + m;
        v2f bl; bl.x = blp[0]; bl.y = blp[D_];
        v2f br; br.x = brp[0]; br.y = brp[D_];

        accL = __builtin_amdgcn_wmma_f32_16x16x4_f32(false, a, false, bl,
                                                     (short)0, accL, false, false);
        accR = __builtin_amdgcn_wmma_f32_16x16x4_f32(false, a, false, br,
                                                     (short)0, accR, false, false);
    }

#pragma unroll
    for (int v = 0; v < 8; ++v) {
        const size_t o = (size_t)(row0 + v + half * 8) * D_ + col0 + m;
        gl[o] = accL[v];
        gr[o] = accR[v];
    }
}

// ---------------------------------------------------------------------------
// Stage 1b: el[row] = gl[row,:]·a, er[row] = gr[row,:]·a   (B*L rows)
// Tiny: 2 MB of traffic total.
// ---------------------------------------------------------------------------
__global__ void rowdot_kernel(const float* __restrict__ gl,
                              const float* __restrict__ gr,
                              const float* __restrict__ a,
                              float* __restrict__ el,
                              float* __restrict__ er)
{
    __shared__ float as[D_];
    const int tid = threadIdx.x;
    if (tid < D_) as[tid] = a[tid];
    __syncthreads();

    const int row = blockIdx.x * blockDim.x + tid;
    if (row >= B_ * L_) return;
    const float* pl = gl + (size_t)row * D_;
    const float* pr = gr + (size_t)row * D_;
    float sl = 0.0f, sr = 0.0f;
#pragma unroll 8
    for (int d = 0; d < D_; ++d) {
        sl = fmaf(pl[d], as[d], sl);
        sr = fmaf(pr[d], as[d], sr);
    }
    el[row] = sl;
    er[row] = sr;
}

// ---------------------------------------------------------------------------
// Stage 2: e[b,x,y] = P*(el[y]+er[x]) + Q*sum_d |gl[b,y,d]+gr[b,x,d]|*a[d]
// Dominant VALU stage: inner loop is v_add_f32 + v_fma_f32(|src|) only.
// 32x32 (x,y) tile per 256-thread block, 2x2 micro-tile per thread,
// float2 LDS reads (row stride 66 floats: 8B-aligned, bank-spread).
// ---------------------------------------------------------------------------
#define EPAD 2
__global__ void e_kernel(const float* __restrict__ gl,  // (B, L, D)
                         const float* __restrict__ gr,  // (B, L, D)
                         const float* __restrict__ a,   // (D)
                         const float* __restrict__ el,  // (B*L)
                         const float* __restrict__ er,  // (B*L)
                         float* __restrict__ e)         // (B, L, L)
{
    __shared__ float gls[32][D_ + EPAD];  // y rows
    __shared__ float grs[32][D_ + EPAD];  // x rows
    __shared__ float as[D_];

    const int b   = blockIdx.z;
    const int x0  = blockIdx.y * 32;
    const int y0  = blockIdx.x * 32;
    const int tid = threadIdx.x;  // 256 threads

    for (int i = tid; i < 32 * D_; i += 256) {
        const int r = i >> 6, c = i & 63;
        gls[r][c] = gl[((size_t)b * L_ + y0 + r) * D_ + c];
        grs[r][c] = gr[((size_t)b * L_ + x0 + r) * D_ + c];
    }
    if (tid < D_) as[tid] = a[tid];
    __syncthreads();

    const int tx2 = tid & 15;   // y pair selector
    const int ty2 = tid >> 4;   // x pair selector
    const int ya = 2 * tx2, yb = ya + 1;
    const int xa = 2 * ty2, xb = xa + 1;

    float acc00 = 0.0f, acc01 = 0.0f, acc10 = 0.0f, acc11 = 0.0f;

#pragma unroll 4
    for (int d = 0; d < D_; d += 2) {
        const float2 gy0 = *(const float2*)&gls[ya][d];
        const float2 gy1 = *(const float2*)&gls[yb][d];
        const float2 gx0 = *(const float2*)&grs[xa][d];
        const float2 gx1 = *(const float2*)&grs[xb][d];
        const float2 ad  = *(const float2*)&as[d];

        acc00 = fmaf(fabsf(gy0.x + gx0.x), ad.x, acc00);
        acc01 = fmaf(fabsf(gy1.x + gx0.x), ad.x, acc01);
        acc10 = fmaf(fabsf(gy0.x + gx1.x), ad.x, acc10);
        acc11 = fmaf(fabsf(gy1.x + gx1.x), ad.x, acc11);

        acc00 = fmaf(fabsf(gy0.y + gx0.y), ad.y, acc00);
        acc01 = fmaf(fabsf(gy1.y + gx0.y), ad.y, acc01);
        acc10 = fmaf(fabsf(gy0.y + gx1.y), ad.y, acc10);
        acc11 = fmaf(fabsf(gy1.y + gx1.y), ad.y, acc11);
    }

    const float el0 = el[(size_t)b * L_ + y0 + ya];
    const float el1 = el[(size_t)b * L_ + y0 + yb];
    const float er0 = er[(size_t)b * L_ + x0 + xa];
    const float er1 = er[(size_t)b * L_ + x0 + xb];

    float2 r0, r1;
    r0.x = P_COEF * (el0 + er0) + Q_COEF * acc00;
    r0.y = P_COEF * (el1 + er0) + Q_COEF * acc01;
    r1.x = P_COEF * (el0 + er1) + Q_COEF * acc10;
    r1.y = P_COEF * (el1 + er1) + Q_COEF * acc11;

    *(float2*)&e[((size_t)b * L_ + x0 + xa) * L_ + y0 + ya] = r0;
    *(float2*)&e[((size_t)b * L_ + x0 + xb) * L_ + y0 + ya] = r1;
}

// ---------------------------------------------------------------------------
// Stage 3: in-place row softmax over last dim (L = 1024), one block per row.
// ---------------------------------------------------------------------------
__global__ void softmax_kernel(float* __restrict__ e)  // (B*L, L)
{
    __shared__ float red[256];
    float* p = e + (size_t)blockIdx.x * L_;
    const int tid = threadIdx.x;

    float v[4];
    float m = -INFINITY;
#pragma unroll
    for (int i = 0; i < 4; ++i) {
        v[i] = p[tid + i * 256];
        m = fmaxf(m, v[i]);
    }
    red[tid] = m;
    __syncthreads();
    for (int s = 128; s > 0; s >>= 1) {
        if (tid < s) red[tid] = fmaxf(red[tid], red[tid + s]);
        __syncthreads();
    }
    m = red[0];
    __syncthreads();

    float sum = 0.0f;
#pragma unroll
    for (int i = 0; i < 4; ++i) {
        v[i] = __expf(v[i] - m);
        sum += v[i];
    }
    red[tid] = sum;
    __syncthreads();
    for (int s = 128; s > 0; s >>= 1) {
        if (tid < s) red[tid] += red[tid + s];
        __syncthreads();
    }
    const float inv = 1.0f / red[0];
#pragma unroll
    for (int i = 0; i < 4; ++i) p[tid + i * 256] = v[i] * inv;
}

// ---------------------------------------------------------------------------
// Stage 4: y[b,x,d] = sum_k scores[b,x,k] * gr[b,k,d]   (fp32 WMMA 16x16x4)
// ---------------------------------------------------------------------------
__global__ void out_gemm_kernel(const float* __restrict__ s,   // (B, L, L)
                                const float* __restrict__ gr,  // (B, L, D)
                                float* __restrict__ y)         // (B, L, D)
{
    const int tile = blockIdx.x;
    const int dt = tile % (D_ / 16);
    const int xt = (tile / (D_ / 16)) % (L_ / 16);
    const int b  = tile / ((D_ / 16) * (L_ / 16));
    const int x0 = xt * 16;
    const int d0 = dt * 16;

    const int lane = threadIdx.x;
    const int m    = lane & 15;
    const int half = lane >> 4;

    const float* srow  = s + ((size_t)b * L_ + x0 + m) * L_;
    const float* gbase = gr + (size_t)b * L_ * D_ + d0 + m;

    v8f acc = {};
#pragma unroll 4
    for (int k0 = 0; k0 < L_; k0 += 4) {
        const int k = k0 + half * 2;
        v2f a; a.x = srow[k]; a.y = srow[k + 1];
        const float* bp = gbase + (size_t)k * D_;
        v2f bb; bb.x = bp[0]; bb.y = bp[D_];
        acc = __builtin_amdgcn_wmma_f32_16x16x4_f32(false, a, false, bb,
                                                    (short)0, acc, false, false);
    }

#pragma unroll
    for (int v = 0; v < 8; ++v)
        y[((size_t)b * L_ + x0 + v + half * 8) * D_ + d0 + m] = acc[v];
}

// ---------------------------------------------------------------------------
extern "C" void kernel_launch(void* const* d_in, const int* in_sizes, int n_in,
                              void* d_out, int out_size, void* d_ws, size_t ws_size,
                              hipStream_t stream) {
    const float* q  = (const float*)d_in[0];  // queries (B,L,C)
    // d_in[1] = keys, d_in[2] = values: unused by the reference math
    const float* wl = (const float*)d_in[3];  // linear_l (C,D)
    const float* wr = (const float*)d_in[4];  // linear_r (C,D)
    const float* a  = (const float*)d_in[5];  // a (D,1)

    float* out    = (float*)d_out;
    float* y      = out;                                  // (B,L,D)
    float* scores = out + (size_t)B_ * L_ * D_;           // (B,L,L)

    float* gl = (float*)d_ws;                             // (B*L, D)
    float* gr = gl + (size_t)B_ * L_ * D_;                // (B*L, D)
    float* el = gr + (size_t)B_ * L_ * D_;                // (B*L)
    float* er = el + (size_t)B_ * L_;                     // (B*L)

    // Stage 1: projections (WMMA f32)
    proj_kernel<<<(B_ * L_ / 16) * (D_ / 16), 32, 0, stream>>>(q, wl, wr, gl, gr);

    // Stage 1b: row dots el = gl·a, er = gr·a
    rowdot_kernel<<<(B_ * L_ + 255) / 256, 256, 0, stream>>>(gl, gr, a, el, er);

    // Stage 2: pre-softmax logits straight into the scores output slot
    e_kernel<<<dim3(L_ / 32, L_ / 32, B_), 256, 0, stream>>>(gl, gr, a, el, er, scores);

    // Stage 3: in-place softmax
    softmax_kernel<<<B_ * L_, 256, 0, stream>>>(scores);

    // Stage 4: y = scores @ g_r (WMMA f32)
    out_gemm_kernel<<<B_ * (L_ / 16) * (D_ / 16), 32, 0, stream>>>(scores, gr, y);
}